// KGM_55370718380663
// MI455X (gfx1250) — compile-verified
//
#include <hip/hip_runtime.h>
#include <math.h>

typedef float v8f __attribute__((ext_vector_type(8)));
typedef float v2f __attribute__((ext_vector_type(2)));

#define Nn 4096
#define Cc 128
#define Kn 32
#define R1 5
#define CONV_R2 0.04f
#define SHADOW_INF 1000000.0f

static __device__ __forceinline__ v8f wmma_f32(v2f a, v2f b, v8f c) {
    return __builtin_amdgcn_wmma_f32_16x16x4_f32(false, a, false, b, (short)0, c, false, false);
}

// ---------------------------------------------------------------------------
// Kernel 1: ball query (first K in-range by index) + kernel-point correlation
// One wave32 per point. idx: (B*N, 32) int. wcorr: (B*N, 5, 32) float.
// ---------------------------------------------------------------------------
__global__ void ballquery_corr(const float* __restrict__ pos,
                               const float* __restrict__ qp,
                               int* __restrict__ idx,
                               float* __restrict__ wcorr) {
    int wid  = (blockIdx.x * blockDim.x + threadIdx.x) >> 5;   // point id in [0, B*N)
    int lane = threadIdx.x & 31;
    int b = wid >> 12;
    int n = wid & (Nn - 1);
    const float* pb = pos + (size_t)b * 3 * Nn;
    float px = pb[n], py = pb[Nn + n], pz = pb[2 * Nn + n];
    float pn = px * px + py * py + pz * pz;
    int* ib = idx + (size_t)wid * Kn;
    int cnt = 0;
    for (int chunk = 0; chunk < Nn / 32; ++chunk) {
        int m = chunk * 32 + lane;
        float mx = pb[m], my = pb[Nn + m], mz = pb[2 * Nn + m];
        float mn = mx * mx + my * my + mz * mz;
        float d2 = pn + mn - 2.0f * (px * mx + py * my + pz * mz);
        bool inr = (d2 <= CONV_R2);
        unsigned mask = __builtin_amdgcn_ballot_w32(inr);
        int slot = cnt + __builtin_popcount(mask & ((1u << lane) - 1u));
        if (inr && slot < Kn) ib[slot] = m;
        cnt += __builtin_popcount(mask);
        if (cnt >= Kn) break;
    }
    int mincnt = cnt < Kn ? cnt : Kn;
    if (lane >= mincnt) ib[lane] = Nn;                          // shadow pad
    // correlation weights: one lane per neighbor slot
    int m = ib[lane];
    float nx, ny, nz;
    if (m == Nn) { nx = ny = nz = SHADOW_INF; }
    else         { nx = pb[m]; ny = pb[Nn + m]; nz = pb[2 * Nn + m]; }
    float rx = nx - px, ry = ny - py, rz = nz - pz;
    float* wb = wcorr + (size_t)wid * R1 * Kn;
    for (int r = 0; r < R1; ++r) {
        float dx = rx - qp[0 * R1 + r];
        float dy = ry - qp[1 * R1 + r];
        float dz = rz - qp[2 * R1 + r];
        float s  = sqrtf(dx * dx + dy * dy + dz * dz);
        float w  = 1.0f - s * 10.0f;                            // 1 - s/0.1
        wb[r * Kn + lane] = w > 0.0f ? w : 0.0f;
    }
}

// ---------------------------------------------------------------------------
// Kernel 2: feat0 = w_begin @ feat + b_begin  -> (B, N+1, C), row N zeroed
// grid = B*(N/16), block 256 (8 waves, one 16-out-channel tile each)
// ---------------------------------------------------------------------------
__global__ void gemm_begin(const float* __restrict__ feat,
                           const float* __restrict__ Wb,
                           const float* __restrict__ bias,
                           float* __restrict__ feat0) {
    int blk = blockIdx.x;
    int b   = blk >> 8;                     // N/16 = 256
    int n0  = (blk & 255) * 16;
    __shared__ float Fs[Cc * 16];
    int t = threadIdx.x;
    const float* fb = feat + (size_t)b * Cc * Nn;
    for (int e = t; e < Cc * 16; e += 256) {
        int c = e >> 4, nn = e & 15;
        Fs[e] = fb[c * Nn + n0 + nn];
    }
    __syncthreads();
    int w = t >> 5, lane = t & 31;
    int o0 = w * 16;
    int m = lane & 15, hi = lane >> 4, kp2 = hi * 2;
    v8f acc = {};
    for (int k0 = 0; k0 < Cc; k0 += 4) {
        v2f a, bb;
        a.x  = Wb[(o0 + m) * Cc + k0 + kp2];
        a.y  = Wb[(o0 + m) * Cc + k0 + kp2 + 1];
        bb.x = Fs[(k0 + kp2) * 16 + m];
        bb.y = Fs[(k0 + kp2 + 1) * 16 + m];
        acc  = wmma_f32(a, bb, acc);
    }
    float* ob = feat0 + (size_t)b * (Nn + 1) * Cc + (size_t)(n0 + m) * Cc + o0 + 8 * hi;
    for (int v = 0; v < 8; ++v) ob[v] = acc[v] + bias[o0 + 8 * hi + v];
}

// ---------------------------------------------------------------------------
// Kernel 3: wf[bn][r][c] = sum_k wcorr[bn][r][k] * feat0[b][idx[bn][k]][c]
// One wave per point; lane owns 4 channels.
// ---------------------------------------------------------------------------
__global__ void wf_kernel(const float* __restrict__ feat0,
                          const int* __restrict__ idx,
                          const float* __restrict__ wcorr,
                          float* __restrict__ wf) {
    int wid  = (blockIdx.x * blockDim.x + threadIdx.x) >> 5;
    int lane = threadIdx.x & 31;
    int b = wid >> 12;
    const int*   ib = idx + (size_t)wid * Kn;
    const float* wb = wcorr + (size_t)wid * R1 * Kn;
    const float* fb = feat0 + (size_t)b * (Nn + 1) * Cc;
    float acc[R1][4] = {};
    for (int k = 0; k < Kn; ++k) {
        int m = ib[k];                                    // shadow m==N -> zero row
        float4 f = *(const float4*)(fb + (size_t)m * Cc + lane * 4);
        float fe[4] = {f.x, f.y, f.z, f.w};
        for (int r = 0; r < R1; ++r) {
            float w = wb[r * Kn + k];
            for (int c = 0; c < 4; ++c) acc[r][c] += w * fe[c];
        }
    }
    float* ob = wf + (size_t)wid * R1 * Cc;
    for (int r = 0; r < R1; ++r)
        for (int c = 0; c < 4; ++c)
            ob[r * Cc + lane * 4 + c] = acc[r][c];
}

// ---------------------------------------------------------------------------
// Kernel 4: t1[col][d] = relu( sum_K wf[col][K] * kpw[K][d] + kpb[d] ), K=640
// grid = B*N/16, block 256
// ---------------------------------------------------------------------------
__global__ void gemm_kp(const float* __restrict__ wf,
                        const float* __restrict__ kpw,
                        const float* __restrict__ kpb,
                        float* __restrict__ t1) {
    int col0 = blockIdx.x * 16;
    int t = threadIdx.x;
    int w = t >> 5, lane = t & 31;
    int d0 = w * 16;
    int m = lane & 15, hi = lane >> 4, kp2 = hi * 2;
    const float* wrow = wf + (size_t)(col0 + m) * (R1 * Cc);
    v8f acc = {};
    for (int k0 = 0; k0 < R1 * Cc; k0 += 4) {
        v2f a, bb;
        a.x  = kpw[(k0 + kp2) * Cc + d0 + m];
        a.y  = kpw[(k0 + kp2 + 1) * Cc + d0 + m];
        bb.x = wrow[k0 + kp2];
        bb.y = wrow[k0 + kp2 + 1];
        acc  = wmma_f32(a, bb, acc);
    }
    float* ob = t1 + (size_t)(col0 + m) * Cc + d0 + 8 * hi;
    for (int v = 0; v < 8; ++v) {
        float x = acc[v] + kpb[d0 + 8 * hi + v];
        ob[v] = x > 0.0f ? x : 0.0f;
    }
}

// ---------------------------------------------------------------------------
// Kernel 5: qf2[col][o] = relu( w_end @ t1 + b_end + feat[b][o][n] )
// ---------------------------------------------------------------------------
__global__ void gemm_end(const float* __restrict__ t1,
                         const float* __restrict__ We,
                         const float* __restrict__ be,
                         const float* __restrict__ feat,
                         float* __restrict__ qf2) {
    int col0 = blockIdx.x * 16;
    int t = threadIdx.x;
    int w = t >> 5, lane = t & 31;
    int o0 = w * 16;
    int m = lane & 15, hi = lane >> 4, kp2 = hi * 2;
    const float* trow = t1 + (size_t)(col0 + m) * Cc;
    v8f acc = {};
    for (int k0 = 0; k0 < Cc; k0 += 4) {
        v2f a, bb;
        a.x  = We[(o0 + m) * Cc + k0 + kp2];
        a.y  = We[(o0 + m) * Cc + k0 + kp2 + 1];
        bb.x = trow[k0 + kp2];
        bb.y = trow[k0 + kp2 + 1];
        acc  = wmma_f32(a, bb, acc);
    }
    int col = col0 + m;
    int b = col >> 12, n = col & (Nn - 1);
    float* ob = qf2 + (size_t)col * Cc;
    for (int v = 0; v < 8; ++v) {
        int o = o0 + 8 * hi + v;
        float x = acc[v] + be[o] + feat[(size_t)b * Cc * Nn + (size_t)o * Nn + n];
        ob[o] = x > 0.0f ? x : 0.0f;
    }
}

// ---------------------------------------------------------------------------
// Kernel 6: fused 2-layer MLP over 65536 columns (b,n,r), 16 cols per block
// cat rows: [0,128)=wf[r+1], [128,131)=qp, [131,259)=qf2, [259,387)=feat, 387=0
// ---------------------------------------------------------------------------
__global__ void mlp_kernel(const float* __restrict__ wf,
                           const float* __restrict__ qp,
                           const float* __restrict__ qf2,
                           const float* __restrict__ feat,
                           const float* __restrict__ wp,   // (256, 388) padded
                           const float* __restrict__ b1,
                           const float* __restrict__ w2,   // (128, 256)
                           const float* __restrict__ b2,
                           float* __restrict__ out) {
    __shared__ float catT[388 * 16];
    __shared__ float hT[256 * 16];
    int colbase = blockIdx.x * 16;
    int t = threadIdx.x;
    for (int e = t; e < 388 * 16; e += 256) {
        int row = e >> 4, nn = e & 15;
        int col = colbase + nn;
        int b = col >> 14, inb = col & 16383;
        int n = inb >> 2, r = inb & 3;
        float v;
        if (row < 128)      v = wf[((size_t)((b << 12) + n) * R1 + r + 1) * Cc + row];
        else if (row < 131) v = qp[(row - 128) * R1 + r + 1];
        else if (row < 259) v = qf2[(size_t)((b << 12) + n) * Cc + row - 131];
        else if (row < 387) v = feat[(size_t)b * Cc * Nn + (size_t)(row - 259) * Nn + n];
        else                v = 0.0f;
        catT[e] = v;
    }
    __syncthreads();
    int w = t >> 5, lane = t & 31;
    int m = lane & 15, hi = lane >> 4, kp2 = hi * 2;
    // GEMM1: M=256, K=388 (two 16-row tiles per wave)
    int o0 = w * 32;
    v8f acc0 = {}, acc1 = {};
    for (int k0 = 0; k0 < 388; k0 += 4) {
        v2f bb, a0, a1;
        bb.x = catT[(k0 + kp2) * 16 + m];
        bb.y = catT[(k0 + kp2 + 1) * 16 + m];
        a0.x = wp[(o0 + m) * 388 + k0 + kp2];
        a0.y = wp[(o0 + m) * 388 + k0 + kp2 + 1];
        a1.x = wp[(o0 + 16 + m) * 388 + k0 + kp2];
        a1.y = wp[(o0 + 16 + m) * 388 + k0 + kp2 + 1];
        acc0 = wmma_f32(a0, bb, acc0);
        acc1 = wmma_f32(a1, bb, acc1);
    }
    for (int v = 0; v < 8; ++v) {
        int o = o0 + 8 * hi + v;
        float x = acc0[v] + b1[o];
        hT[o * 16 + m] = x > 0.0f ? x : 0.0f;
        float y = acc1[v] + b1[o + 16];
        hT[(o + 16) * 16 + m] = y > 0.0f ? y : 0.0f;
    }
    __syncthreads();
    // GEMM2: M=128, K=256
    int d0 = w * 16;
    v8f acc = {};
    for (int k0 = 0; k0 < 256; k0 += 4) {
        v2f a, bb;
        a.x  = w2[(d0 + m) * 256 + k0 + kp2];
        a.y  = w2[(d0 + m) * 256 + k0 + kp2 + 1];
        bb.x = hT[(k0 + kp2) * 16 + m];
        bb.y = hT[(k0 + kp2 + 1) * 16 + m];
        acc  = wmma_f32(a, bb, acc);
    }
    int col = colbase + m;
    int b = col >> 14, inb = col & 16383;
    float* ob = out + 12 + (size_t)b * (Cc * 16384) + inb;   // queries (B, TD, N, R)
    for (int v = 0; v < 8; ++v) {
        int d = d0 + 8 * hi + v;
        float x = acc[v] + b2[d];
        ob[(size_t)d * 16384] = x > 0.0f ? x : 0.0f;
    }
}

// ---------------------------------------------------------------------------
// Prep: pad w_mlp1 (256x387 -> 256x388 with zero col), zero feat0 shadow rows
// ---------------------------------------------------------------------------
__global__ void prep_kernel(const float* __restrict__ w_mlp1,
                            float* __restrict__ wp,
                            float* __restrict__ feat0) {
    int t = blockIdx.x * blockDim.x + threadIdx.x;
    if (t < 256 * 388) {
        int row = t / 388, col = t % 388;
        wp[t] = (col < 387) ? w_mlp1[row * 387 + col] : 0.0f;
    }
    if (t < 4 * Cc) {
        int b = t >> 7, c = t & (Cc - 1);
        feat0[(size_t)b * (Nn + 1) * Cc + (size_t)Nn * Cc + c] = 0.0f;
    }
}

// First tuple output: query_pos[0, :, 0, 1:] -> 12 floats (3,4)
__global__ void out_qp_kernel(const float* __restrict__ qp, float* __restrict__ out) {
    int t = threadIdx.x;
    if (t < 12) {
        int c = t >> 2, r = t & 3;
        out[t] = qp[c * R1 + r + 1];
    }
}

extern "C" void kernel_launch(void* const* d_in, const int* in_sizes, int n_in,
                              void* d_out, int out_size, void* d_ws, size_t ws_size,
                              hipStream_t stream) {
    (void)in_sizes; (void)n_in; (void)out_size; (void)ws_size;
    const float* pos     = (const float*)d_in[0];
    const float* feat    = (const float*)d_in[1];
    const float* qp      = (const float*)d_in[2];
    const float* w_begin = (const float*)d_in[3];
    const float* b_begin = (const float*)d_in[4];
    const float* kpw     = (const float*)d_in[5];
    const float* kpb     = (const float*)d_in[6];
    const float* w_end   = (const float*)d_in[7];
    const float* b_end   = (const float*)d_in[8];
    const float* w_mlp1  = (const float*)d_in[9];
    const float* b_mlp1  = (const float*)d_in[10];
    const float* w_mlp2  = (const float*)d_in[11];
    const float* b_mlp2  = (const float*)d_in[12];
    float* out = (float*)d_out;

    char* ws = (char*)d_ws;
    int*   idx   = (int*)  (ws + 0);                     // B*N*K ints   (2 MB)
    float* wcorr = (float*)(ws + 2097152);               // B*N*5*K      (10.5 MB)
    float* feat0 = (float*)(ws + 12582912);              // B*(N+1)*C    (8.4 MB)
    float* wf    = (float*)(ws + 20973568);              // B*N*5*C      (42 MB)
    float* t1    = (float*)(ws + 62916608);              // B*N*C        (8.4 MB)
    float* qf2   = (float*)(ws + 71305216);              // B*N*C        (8.4 MB)
    float* wp    = (float*)(ws + 79693824);              // 256*388      (0.4 MB)

    prep_kernel<<<388, 256, 0, stream>>>(w_mlp1, wp, feat0);
    out_qp_kernel<<<1, 32, 0, stream>>>(qp, out);
    ballquery_corr<<<2048, 256, 0, stream>>>(pos, qp, idx, wcorr);   // 16384 waves
    gemm_begin<<<1024, 256, 0, stream>>>(feat, w_begin, b_begin, feat0);
    wf_kernel<<<2048, 256, 0, stream>>>(feat0, idx, wcorr, wf);
    gemm_kp<<<1024, 256, 0, stream>>>(wf, kpw, kpb, t1);
    gemm_end<<<1024, 256, 0, stream>>>(t1, w_end, b_end, feat, qf2);
    mlp_kernel<<<4096, 256, 0, stream>>>(wf, qp, qf2, feat, wp, b_mlp1, w_mlp2, b_mlp2, out);
}